// JointLocationLoss_15410342658542
// MI455X (gfx1250) — compile-verified
//
#include <hip/hip_runtime.h>
#include <hip/hip_bf16.h>
#include <stdint.h>

#define BN 32
#define JN 18
#define NPAIR (BN * JN)          // 576
#define VOX (64 * 64 * 64)       // 262144 voxels per (b,j)
#define NF4 (VOX / 4)            // 65536 float4 per (b,j)
#define TPB 256
#define NWAVE (TPB / 32)
#define NSTAGE (NF4 / TPB)       // 256 stages of 1024 floats
#define DEPTH 8                  // async pipeline depth
#define L2E 1.4426950408889634f  // log2(e)

__device__ __forceinline__ float fexp2(float x) {
  return __builtin_amdgcn_exp2f(x);  // v_exp_f32
}

// Issue one 16B/lane async copy global -> LDS (CDNA5 async-tensor path).
__device__ __forceinline__ void async_copy_b128(uint32_t lds_off, const float4* gaddr) {
  asm volatile("global_load_async_to_lds_b128 %0, %1, off"
               :: "v"(lds_off), "v"(gaddr) : "memory");
}

__global__ __launch_bounds__(TPB) void k_softargmax_loss(
    const float* __restrict__ hm,        // [32,18,64,64,64]
    const float* __restrict__ gt_coord,  // [32,18,3]
    const float* __restrict__ gt_vis,    // [32,18]
    const float* __restrict__ gt_hd,     // [32]
    float* __restrict__ pair_loss)       // [576]
{
  __shared__ float4 sbuf[NWAVE][DEPTH][32];  // 32 KB staging
  __shared__ float  wred[NWAVE][5];

  const int tid  = threadIdx.x;
  const int lane = tid & 31;
  const int wv   = tid >> 5;
  const int pair = blockIdx.x;

  const float4* src = (const float4*)hm + (size_t)pair * NF4 + tid;

  // Low 32 bits of the flat pointer == LDS byte offset (aperture in [63:32]).
  const uint32_t lds_lane_base = (uint32_t)(uintptr_t)(&sbuf[wv][0][lane]);

  // ---- prologue: fill the pipeline ----
#pragma unroll
  for (int d = 0; d < DEPTH; ++d)
    async_copy_b128(lds_lane_base + (uint32_t)d * 512u, src + d * TPB);

  // online-softmax state (log2 domain: t = v*log2(e))
  float m = -1e30f, s = 0.f, sxf = 0.f, sy = 0.f, sz = 0.f;
  const float xbf  = (float)((tid * 4) & 63);  // x base, constant per lane
  const int   row0 = tid >> 4;                 // row = i*16 + row0

  auto accum = [&](float4 v, int row) {
    float t0 = v.x * L2E, t1 = v.y * L2E, t2 = v.z * L2E, t3 = v.w * L2E;
    float tm = fmaxf(fmaxf(t0, t1), fmaxf(t2, t3));
    if (tm > m) {  // rare: rescale running sums
      float c = fexp2(m - tm);
      s *= c; sxf *= c; sy *= c; sz *= c;
      m = tm;
    }
    float e0 = fexp2(t0 - m), e1 = fexp2(t1 - m);
    float e2 = fexp2(t2 - m), e3 = fexp2(t3 - m);
    float es    = (e0 + e1) + (e2 + e3);
    float inner = __builtin_fmaf(3.f, e3, __builtin_fmaf(2.f, e2, e1));
    s   += es;
    sxf += inner;
    sy = __builtin_fmaf(es, (float)(row & 63), sy);
    sz = __builtin_fmaf(es, (float)(row >> 6), sz);
  };

  // ---- main loop: consume stage i, reissue stage i+DEPTH ----
  for (int i = 0; i < NSTAGE - DEPTH; ++i) {
    asm volatile("s_wait_asynccnt 0x7" ::: "memory");  // oldest stage landed in LDS
    const int slot = i & (DEPTH - 1);
    float4 v = sbuf[wv][slot][lane];                   // ds_load_b128
    asm volatile("s_wait_dscnt 0x0" ::: "memory");     // data in VGPRs before reuse of slot
    async_copy_b128(lds_lane_base + (uint32_t)slot * 512u, src + (i + DEPTH) * TPB);
    accum(v, i * 16 + row0);
  }
  // ---- drain ----
  asm volatile("s_wait_asynccnt 0x0" ::: "memory");
  for (int i = NSTAGE - DEPTH; i < NSTAGE; ++i) {
    float4 v = sbuf[wv][i & (DEPTH - 1)][lane];
    accum(v, i * 16 + row0);
  }

  // fold constant-x part, then wave32 reduction of online states
  float sx = __builtin_fmaf(s, xbf, sxf);
  float M = m;
#pragma unroll
  for (int off = 16; off >= 1; off >>= 1)
    M = fmaxf(M, __shfl_xor(M, off, 32));
  float c = fexp2(m - M);
  s *= c; sx *= c; sy *= c; sz *= c;
#pragma unroll
  for (int off = 16; off >= 1; off >>= 1) {
    s  += __shfl_xor(s,  off, 32);
    sx += __shfl_xor(sx, off, 32);
    sy += __shfl_xor(sy, off, 32);
    sz += __shfl_xor(sz, off, 32);
  }
  if (lane == 0) {
    wred[wv][0] = M; wred[wv][1] = s; wred[wv][2] = sx;
    wred[wv][3] = sy; wred[wv][4] = sz;
  }
  __syncthreads();

  if (tid == 0) {
    float gm = wred[0][0];
#pragma unroll
    for (int w = 1; w < NWAVE; ++w) gm = fmaxf(gm, wred[w][0]);
    float S = 0.f, SX = 0.f, SY = 0.f, SZ = 0.f;
#pragma unroll
    for (int w = 0; w < NWAVE; ++w) {
      float cc = fexp2(wred[w][0] - gm);
      S  += wred[w][1] * cc; SX += wred[w][2] * cc;
      SY += wred[w][3] * cc; SZ += wred[w][4] * cc;
    }
    float inv = 1.0f / S;
    float ax = SX * inv, ay = SY * inv, az = SZ * inv;  // already 0-indexed
    const float* g = gt_coord + pair * 3;
    float vis = gt_vis[pair];
    float hd  = gt_hd[pair / JN];
    float loss = vis * (fabsf(ax - g[0]) + fabsf(ay - g[1]) + hd * fabsf(az - g[2]))
               * (1.0f / 3.0f);
    pair_loss[pair] = loss;
  }
}

__global__ __launch_bounds__(NPAIR) void k_final(const float* __restrict__ pl,
                                                 float* __restrict__ out) {
  __shared__ float red[NPAIR];
  const int t = threadIdx.x;
  red[t] = pl[t];
  __syncthreads();
#pragma unroll
  for (int off = 512; off >= 1; off >>= 1) {
    if (t < off && t + off < NPAIR) red[t] += red[t + off];
    __syncthreads();
  }
  if (t == 0) out[0] = red[0] * (1.0f / NPAIR);
}

extern "C" void kernel_launch(void* const* d_in, const int* in_sizes, int n_in,
                              void* d_out, int out_size, void* d_ws, size_t ws_size,
                              hipStream_t stream) {
  const float* hm       = (const float*)d_in[0];
  const float* gt_coord = (const float*)d_in[1];
  const float* gt_vis   = (const float*)d_in[2];
  const float* gt_hd    = (const float*)d_in[3];
  float* pair_loss = (float*)d_ws;   // 576 floats of scratch
  float* out       = (float*)d_out;

  k_softargmax_loss<<<NPAIR, TPB, 0, stream>>>(hm, gt_coord, gt_vis, gt_hd, pair_loss);
  k_final<<<1, NPAIR, 0, stream>>>(pair_loss, out);
}